// QuantGraphPooling_90434831384775
// MI455X (gfx1250) — compile-verified
//
#include <hip/hip_runtime.h>
#include <hip/hip_bf16.h>

// QuantGraphPooling for MI455X (gfx1250, wave32).
// Memory/atomic-bound pipeline; uses gfx1250 async global->LDS copy
// (global_load_async_to_lds_b32 / s_wait_asynccnt) and global_prefetch_b8.

#define POOLK  4
#define GRIDD  64
#define G3C    262144   // 64^3

typedef unsigned int u32;

// ---- monotonic float<->uint ordering (for atomic max on floats) ----
__device__ __forceinline__ u32 f2o(float f) {
    u32 b = __float_as_uint(f);
    return (b & 0x80000000u) ? ~b : (b | 0x80000000u);
}
__device__ __forceinline__ float o2f(u32 o) {
    return (o & 0x80000000u) ? __uint_as_float(o & 0x7fffffffu)
                             : __uint_as_float(~o);
}

// ---------------- generic helpers ----------------
__global__ void fill_u32_k(u32* __restrict__ p, u32 v, long long n) {
    long long i = (long long)blockIdx.x * blockDim.x + threadIdx.x;
    long long stride = (long long)gridDim.x * blockDim.x;
    for (; i < n; i += stride) p[i] = v;
}

__global__ void copy_u32_k(const u32* __restrict__ s, u32* __restrict__ d, int n) {
    int i = blockIdx.x * 256 + threadIdx.x;
    if (i < n) d[i] = s[i];
}

// ---------------- exclusive scan over G3 elements (3 kernels) ----------------
__global__ void scan_reduce_k(const u32* __restrict__ src, u32* __restrict__ bs, int n) {
    __shared__ u32 sm[256];
    int i = blockIdx.x * 256 + threadIdx.x;
    sm[threadIdx.x] = (i < n) ? src[i] : 0u;
    __syncthreads();
    for (int off = 128; off > 0; off >>= 1) {
        if ((int)threadIdx.x < off) sm[threadIdx.x] += sm[threadIdx.x + off];
        __syncthreads();
    }
    if (threadIdx.x == 0) bs[blockIdx.x] = sm[0];
}

__global__ void scan_tail_k(u32* __restrict__ bs, int nb, u32* __restrict__ total) {
    u32 acc = 0;
    for (int i = 0; i < nb; ++i) { u32 t = bs[i]; bs[i] = acc; acc += t; }
    *total = acc;
}

__global__ void scan_apply_k(const u32* __restrict__ src, const u32* __restrict__ boffs,
                             u32* __restrict__ dst, int n) {
    __shared__ u32 tmp[256];
    int i = blockIdx.x * 256 + threadIdx.x;
    u32 v = (i < n) ? src[i] : 0u;
    tmp[threadIdx.x] = v;
    __syncthreads();
    for (int off = 1; off < 256; off <<= 1) {
        u32 t = ((int)threadIdx.x >= off) ? tmp[threadIdx.x - off] : 0u;
        __syncthreads();
        tmp[threadIdx.x] += t;
        __syncthreads();
    }
    if (i < n) dst[i] = (tmp[threadIdx.x] - v) + boffs[blockIdx.x];
}

// ---------------- stage 1: voxel unique ----------------
__device__ __forceinline__ int voxel_key(const int* __restrict__ verts, int p) {
    int x = verts[3 * p] >> 2, y = verts[3 * p + 1] >> 2, z = verts[3 * p + 2] >> 2;
    return (x << 12) | (y << 6) | z;   // (x*64 + y)*64 + z, all in [0,64)
}

__global__ void mark_present_k(const int* __restrict__ verts, u32* __restrict__ pres, int n) {
    int p = blockIdx.x * 256 + threadIdx.x;
    if (p >= n) return;
    pres[voxel_key(verts, p)] = 1u;    // idempotent store, race-free
}

__global__ void inv_k(const int* __restrict__ verts, const u32* __restrict__ rank,
                      u32* __restrict__ inv, int n) {
    int p = blockIdx.x * 256 + threadIdx.x;
    if (p >= n) return;
    inv[p] = rank[voxel_key(verts, p)];
}

__global__ void pos_scatter_k(const u32* __restrict__ pres, const u32* __restrict__ rank,
                              float* __restrict__ outPos) {
    int k = blockIdx.x * 256 + threadIdx.x;   // grid covers exactly G3
    if (pres[k]) {
        u32 s = rank[k];
        outPos[3 * (size_t)s]     = (float)(k >> 12);
        outPos[3 * (size_t)s + 1] = (float)((k >> 6) & 63);
        outPos[3 * (size_t)s + 2] = (float)(k & 63);
    }
}

__global__ void validv_k(const u32* __restrict__ totals, float* __restrict__ outVV) {
    int s = blockIdx.x * 256 + threadIdx.x;
    outVV[s] = ((u32)s < totals[0]) ? 1.0f : 0.0f;
}

// ---------------- stage 2: segment max pooling ----------------
// One thread per (point, feature). Feature tile is staged into LDS through the
// gfx1250 async copy engine (bypasses VGPRs), then atomicMax on ordered uints.
__global__ void pool_k(const float* __restrict__ feats, const u32* __restrict__ inv,
                       u32* __restrict__ poolU, int total) {
    __shared__ float sfeat[256];
    int g  = blockIdx.x * 256 + threadIdx.x;
    int gc = (g < total) ? g : (total - 1);          // clamp async address in-bounds
    const float* gp = feats + gc;
    u32 ldsOff = (u32)(unsigned long long)(&sfeat[threadIdx.x]);  // low 32b of flat = LDS offset
    asm volatile("global_load_async_to_lds_b32 %0, %1, off"
                 :: "v"(ldsOff), "v"(gp) : "memory");
    asm volatile("s_wait_asynccnt 0" ::: "memory");
    if (g < total) {
        float f = sfeat[threadIdx.x];
        int p = g >> 5, c = g & 31;
        atomicMax(&poolU[((size_t)inv[p] << 5) + c], f2o(f));
    }
}

__global__ void pool_fin_k(u32* __restrict__ poolU, const u32* __restrict__ totals) {
    int i = blockIdx.x * 256 + threadIdx.x;          // grid covers 32*G3
    u32 Nu = totals[0];
    u32 s = (u32)i >> 5;
    u32 u = poolU[i];
    float f = (s < Nu) ? o2f(u) : 0.0f;              // padding segments -> 0 (matches ref)
    ((float*)poolU)[i] = f;                          // in-place decode, same element
}

// ---------------- stage 3: edge remap + dedup ----------------
__global__ void edge_count_k(const int* __restrict__ edges, const u32* __restrict__ inv,
                             u32* __restrict__ bcnt, int n) {
    int e = blockIdx.x * 256 + threadIdx.x;
    if (e >= n) return;
    int pf = e + 8192;
    if (pf < n) __builtin_prefetch(edges + 2 * (size_t)pf, 0, 0);  // global_prefetch_b8
    u32 a = inv[edges[2 * (size_t)e]];
    u32 b = inv[edges[2 * (size_t)e + 1]];
    if (a != b) atomicAdd(&bcnt[a], 1u);             // self-edges dropped (ref zeroes them)
}

__global__ void edge_scatter_k(const int* __restrict__ edges, const u32* __restrict__ inv,
                               u32* __restrict__ bcur, u32* __restrict__ bvals, int n) {
    int e = blockIdx.x * 256 + threadIdx.x;
    if (e >= n) return;
    int pf = e + 8192;
    if (pf < n) __builtin_prefetch(edges + 2 * (size_t)pf, 0, 0);
    u32 a = inv[edges[2 * (size_t)e]];
    u32 b = inv[edges[2 * (size_t)e + 1]];
    if (a != b) bvals[atomicAdd(&bcur[a], 1u)] = b;
}

__global__ void bucket_sort_k(const u32* __restrict__ bstart, const u32* __restrict__ bcnt,
                              u32* __restrict__ bvals, u32* __restrict__ ucnt) {
    int a = blockIdx.x * 256 + threadIdx.x;          // grid covers G3; avg bucket ~8 elems
    u32 n = bcnt[a];
    u32* v = bvals + bstart[a];
    for (u32 i = 1; i < n; ++i) {                    // insertion sort by dest id
        u32 x = v[i];
        int j = (int)i - 1;
        while (j >= 0 && v[j] > x) { v[j + 1] = v[j]; --j; }
        v[j + 1] = x;
    }
    u32 uq = 0, prev = 0xFFFFFFFFu;
    for (u32 i = 0; i < n; ++i) { if (v[i] != prev) { ++uq; prev = v[i]; } }
    ucnt[a] = uq;
}

__global__ void edge_write_k(const u32* __restrict__ bstart, const u32* __restrict__ bcnt,
                             const u32* __restrict__ ustart, const u32* __restrict__ bvals,
                             float* __restrict__ outEdge, float* __restrict__ outVE) {
    int a = blockIdx.x * 256 + threadIdx.x;
    u32 n = bcnt[a];
    if (!n) return;
    const u32* v = bvals + bstart[a];
    u32 o = ustart[a], prev = 0xFFFFFFFFu;
    for (u32 i = 0; i < n; ++i) {
        u32 b = v[i];
        if (b != prev) {
            outEdge[2 * (size_t)o]     = (float)a;
            outEdge[2 * (size_t)o + 1] = (float)b;
            outVE[o] = 1.0f;
            ++o; prev = b;
        }
    }
}

__global__ void selfloop_k(float* __restrict__ outEdge, float* __restrict__ outVE,
                           const u32* __restrict__ totals, int n_edges) {
    int i = blockIdx.x * 256 + threadIdx.x;          // grid covers G3
    size_t r = (size_t)n_edges + (size_t)i;
    outEdge[2 * r]     = (float)i;
    outEdge[2 * r + 1] = (float)i;
    outVE[r] = ((u32)i < totals[0]) ? 1.0f : 0.0f;   // valid_e tail == valid_v
}

// ---------------- host launch ----------------
extern "C" void kernel_launch(void* const* d_in, const int* in_sizes, int n_in,
                              void* d_out, int out_size, void* d_ws, size_t ws_size,
                              hipStream_t stream) {
    const int*   verts = (const int*)d_in[0];
    const float* feats = (const float*)d_in[1];
    const int*   edges = (const int*)d_in[2];
    const int n_pts   = in_sizes[0] / 3;
    const int n_edges = in_sizes[2] / 2;
    const int G3 = G3C;

    // workspace layout (u32 units): ~19.3 MB total
    u32* ws     = (u32*)d_ws;
    u32* rank   = ws;              // G3
    u32* pres   = rank   + G3;     // G3
    u32* bcnt   = pres   + G3;     // G3
    u32* bstart = bcnt   + G3;     // G3
    u32* bcur   = bstart + G3;     // G3
    u32* ucnt   = bcur   + G3;     // G3
    u32* ustart = ucnt   + G3;     // G3
    u32* inv    = ustart + G3;     // n_pts
    u32* bvals  = inv    + n_pts;  // n_edges
    u32* blksum = bvals  + n_edges;// 1024
    u32* totals = blksum + 1024;   // [0]=Nu [1]=keptEdges [2]=Ne

    // output layout (all float32, concatenated in reference return order)
    float* out      = (float*)d_out;
    float* outPos   = out;                                  // [G3,3]
    float* outPool  = outPos  + 3 * (size_t)G3;             // [G3,32]
    u32*   outPoolU = (u32*)outPool;
    float* outEdge  = outPool + 32 * (size_t)G3;            // [nE+G3,2]
    float* outVV    = outEdge + 2 * ((size_t)n_edges + G3); // [G3]
    float* outVE    = outVV   + G3;                         // [nE+G3]

    const int BG3 = G3 / 256;  // 1024 blocks

    // init scratch + outputs (pos/pool/edge contiguous; pool zeros == -inf code)
    fill_u32_k<<<2048, 256, 0, stream>>>(pres, 0u, G3);
    fill_u32_k<<<2048, 256, 0, stream>>>(bcnt, 0u, G3);
    fill_u32_k<<<1, 32, 0, stream>>>(totals, 0u, 8);
    long long nz = 3LL * G3 + 32LL * G3 + 2LL * ((long long)n_edges + G3);
    fill_u32_k<<<4096, 256, 0, stream>>>((u32*)out, 0u, nz);
    fill_u32_k<<<2048, 256, 0, stream>>>((u32*)outVE, 0u, n_edges);

    // stage 1: voxel unique
    mark_present_k<<<(n_pts + 255) / 256, 256, 0, stream>>>(verts, pres, n_pts);
    scan_reduce_k<<<BG3, 256, 0, stream>>>(pres, blksum, G3);
    scan_tail_k<<<1, 1, 0, stream>>>(blksum, BG3, totals + 0);
    scan_apply_k<<<BG3, 256, 0, stream>>>(pres, blksum, rank, G3);
    inv_k<<<(n_pts + 255) / 256, 256, 0, stream>>>(verts, rank, inv, n_pts);
    pos_scatter_k<<<BG3, 256, 0, stream>>>(pres, rank, outPos);
    validv_k<<<BG3, 256, 0, stream>>>(totals, outVV);

    // stage 2: segment-max pooling (async-LDS staged + ordered-uint atomicMax)
    int total = n_pts * 32;
    pool_k<<<(total + 255) / 256, 256, 0, stream>>>(feats, inv, outPoolU, total);
    pool_fin_k<<<(32 * G3) / 256, 256, 0, stream>>>(outPoolU, totals);

    // stage 3: edge dedup (bucket by src, sort by dst, compact unique)
    edge_count_k<<<(n_edges + 255) / 256, 256, 0, stream>>>(edges, inv, bcnt, n_edges);
    scan_reduce_k<<<BG3, 256, 0, stream>>>(bcnt, blksum, G3);
    scan_tail_k<<<1, 1, 0, stream>>>(blksum, BG3, totals + 1);
    scan_apply_k<<<BG3, 256, 0, stream>>>(bcnt, blksum, bstart, G3);
    copy_u32_k<<<BG3, 256, 0, stream>>>(bstart, bcur, G3);
    edge_scatter_k<<<(n_edges + 255) / 256, 256, 0, stream>>>(edges, inv, bcur, bvals, n_edges);
    bucket_sort_k<<<BG3, 256, 0, stream>>>(bstart, bcnt, bvals, ucnt);
    scan_reduce_k<<<BG3, 256, 0, stream>>>(ucnt, blksum, G3);
    scan_tail_k<<<1, 1, 0, stream>>>(blksum, BG3, totals + 2);
    scan_apply_k<<<BG3, 256, 0, stream>>>(ucnt, blksum, ustart, G3);
    edge_write_k<<<BG3, 256, 0, stream>>>(bstart, bcnt, ustart, bvals, outEdge, outVE);
    selfloop_k<<<BG3, 256, 0, stream>>>(outEdge, outVE, totals, n_edges);
}